// GcnLstmModel_68101001446099
// MI455X (gfx1250) — compile-verified
//
#include <hip/hip_runtime.h>
#include <hip/hip_bf16.h>
#include <math.h>

// ---------------------------------------------------------------------------
// GCN(GAT)-LSTM forward for MI455X (gfx1250, wave32).
// - GAT layers & layer-norm are per-timestep => decode head only needs the GAT
//   branch at t = T-1. Only the LSTM needs all T.
// - All GEMMs on V_WMMA_F32_16X16X4_F32 (exact fp32 matrix pipe).
// - LSTM: one persistent kernel; rows are independent across blocks, so the
//   whole T-loop runs with h/c resident in LDS and weights in registers.
// ---------------------------------------------------------------------------

#define Bdim 4
#define Ndim 512
#define Tdim 24
#define NXdim 8
#define Hdim 64
#define PRED 12
#define Mrows (Bdim * Ndim)          // 2048 (b,n) rows
#define XROWS (Bdim * Ndim * Tdim)   // 49152 flat x_h rows

typedef __attribute__((ext_vector_type(2))) float v2f;
typedef __attribute__((ext_vector_type(8))) float v8f;

__device__ inline v8f wmma4(v2f a, v2f b, v8f c) {
  // (neg_a, A, neg_b, B, c_mod, C, reuse_a, reuse_b) -- 8-arg VOP3P pattern
  return __builtin_amdgcn_wmma_f32_16x16x4_f32(false, a, false, b, (short)0, c,
                                               false, false);
}

// Accumulate a 16x16 tile of A(row-major, lda) x B(row-major KxN, ldb), K mult of 4.
// A 16x4 f32: lanes 0-15 row M=l hold K={k0,k0+1}; lanes 16-31 hold K={k0+2,k0+3}
// B 4x16 f32: lanes 0-15 col N=l hold K={k0,k0+1}; lanes 16-31 hold K={k0+2,k0+3}
__device__ inline v8f tile_gemm(const float* __restrict__ A, int lda,
                                const float* __restrict__ Bm, int ldb,
                                int m0, int n0, int K, int lane, v8f c) {
  const int row = m0 + (lane & 15);
  const int col = n0 + (lane & 15);
  const int kh  = (lane >> 4) << 1;
  for (int k0 = 0; k0 < K; k0 += 4) {
    const int ka = k0 + kh;
    v2f a, b;
    a.x = A[row * lda + ka];
    a.y = A[row * lda + ka + 1];
    b.x = Bm[ka * ldb + col];
    b.y = Bm[(ka + 1) * ldb + col];
    c = wmma4(a, b, c);
  }
  return c;
}

// C/D layout: VGPR r holds (M = m0 + r + 8*(lane>=16), N = n0 + (lane&15))
#define TILE_ROW(m0, lane, r) ((m0) + ((lane) >> 4) * 8 + (r))
#define TILE_COL(n0, lane) ((n0) + ((lane) & 15))

// ---------------------------------------------------------------------------
// utility kernels
// ---------------------------------------------------------------------------
__global__ void k_fill(float* __restrict__ p, float v, int n) {
  int i = blockIdx.x * blockDim.x + threadIdx.x;
  if (i < n) p[i] = v;
}

__global__ void k_scatter_adj(const int* __restrict__ ei, int E,
                              float* __restrict__ adj) {
  int k = blockIdx.x * blockDim.x + threadIdx.x;
  if (k < E) {
    int src = ei[k];       // edge_index[0, k]
    int dst = ei[E + k];   // edge_index[1, k]
    adj[src * Ndim + dst] = 0.0f;  // unmasked -> additive bias 0
  }
}

// ---------------------------------------------------------------------------
// x_h = x @ W_fc + b_fc      (49152 x 8) @ (8 x 64)
// ---------------------------------------------------------------------------
__global__ void k_xh(const float* __restrict__ x, const float* __restrict__ Wfc,
                     const float* __restrict__ bfc, float* __restrict__ xh) {
  const int gw = (blockIdx.x * blockDim.x + threadIdx.x) >> 5;
  const int lane = threadIdx.x & 31;
  const int nt = gw & 3;          // 4 col tiles
  const int mt = gw >> 2;         // 3072 row tiles
  const int m0 = mt * 16, n0 = nt * 16;
  v8f c = {};
  c = tile_gemm(x, NXdim, Wfc, Hdim, m0, n0, NXdim, lane, c);
  const int col = TILE_COL(n0, lane);
  const float bias = bfc[col];
  for (int r = 0; r < 8; ++r)
    xh[TILE_ROW(m0, lane, r) * Hdim + col] = c[r] + bias;
}

// ---------------------------------------------------------------------------
// GAT step 1: Hout = A @ W   (2048 x 64) @ (64 x 64); A may be row-strided
// ---------------------------------------------------------------------------
__global__ void k_gat_h(const float* __restrict__ A, int lda,
                        const float* __restrict__ W, float* __restrict__ Hout) {
  const int gw = (blockIdx.x * blockDim.x + threadIdx.x) >> 5;  // 512 waves
  const int lane = threadIdx.x & 31;
  const int nt = gw & 3;
  const int mt = gw >> 2;  // 0..127
  const int m0 = mt * 16, n0 = nt * 16;
  v8f c = {};
  c = tile_gemm(A, lda, W, Hdim, m0, n0, Hdim, lane, c);
  const int col = TILE_COL(n0, lane);
  for (int r = 0; r < 8; ++r)
    Hout[TILE_ROW(m0, lane, r) * Hdim + col] = c[r];
}

// el[m] = H[m,:].a_l ; er[m] = H[m,:].a_r   (one wave per row)
__global__ void k_elr(const float* __restrict__ H, const float* __restrict__ al,
                      const float* __restrict__ ar, float* __restrict__ el,
                      float* __restrict__ er) {
  const int row = (blockIdx.x * blockDim.x + threadIdx.x) >> 5;
  const int lane = threadIdx.x & 31;
  float h0 = H[row * Hdim + lane];
  float h1 = H[row * Hdim + 32 + lane];
  float sl = h0 * al[lane] + h1 * al[32 + lane];
  float sr = h0 * ar[lane] + h1 * ar[32 + lane];
  for (int off = 16; off > 0; off >>= 1) {
    sl += __shfl_xor(sl, off, 32);
    sr += __shfl_xor(sr, off, 32);
  }
  if (lane == 0) { el[row] = sl; er[row] = sr; }
}

// softmax row: alpha[b,m,:] over 512 cols, leaky-relu + adjacency(-inf) bias
__global__ void k_alpha(const float* __restrict__ el, const float* __restrict__ er,
                        const float* __restrict__ adj, float* __restrict__ alpha) {
  const int bm = blockIdx.x;          // 0..2047
  const int b = bm >> 9, m = bm & 511;
  const int tid = threadIdx.x;        // 256 threads, 2 cols each
  const float elm = el[bm];
  float v0, v1;
  {
    float s = elm + er[(b << 9) + tid];
    s = s > 0.0f ? s : 0.2f * s;
    v0 = s + adj[(m << 9) + tid];
    s = elm + er[(b << 9) + tid + 256];
    s = s > 0.0f ? s : 0.2f * s;
    v1 = s + adj[(m << 9) + tid + 256];
  }
  __shared__ float red[256];
  red[tid] = fmaxf(v0, v1);
  __syncthreads();
  for (int s = 128; s > 0; s >>= 1) {
    if (tid < s) red[tid] = fmaxf(red[tid], red[tid + s]);
    __syncthreads();
  }
  const float mx = red[0];            // finite: self-loops guarantee an edge
  __syncthreads();
  float e0 = expf(v0 - mx);
  float e1 = expf(v1 - mx);
  red[tid] = e0 + e1;
  __syncthreads();
  for (int s = 128; s > 0; s >>= 1) {
    if (tid < s) red[tid] += red[tid + s];
    __syncthreads();
  }
  const float inv = 1.0f / red[0];
  float* arow = alpha + (size_t)bm * Ndim;
  arow[tid] = e0 * inv;
  arow[tid + 256] = e1 * inv;
}

// out = gelu(alpha @ H + bias), per batch:  (512x512)@(512x64)
__global__ void k_gat_out(const float* __restrict__ alpha,
                          const float* __restrict__ H,
                          const float* __restrict__ bias,
                          float* __restrict__ gout) {
  const int gw = (blockIdx.x * blockDim.x + threadIdx.x) >> 5;  // 512 waves
  const int lane = threadIdx.x & 31;
  const int b = gw >> 7;              // 128 waves per batch
  const int rem = gw & 127;
  const int m0 = (rem >> 2) * 16;     // 32 row tiles
  const int n0 = (rem & 3) * 16;      // 4 col tiles
  const float* Ab = alpha + (size_t)b * Ndim * Ndim;
  const float* Hb = H + (size_t)b * Ndim * Hdim;
  v8f c = {};
  c = tile_gemm(Ab, Ndim, Hb, Hdim, m0, n0, Ndim, lane, c);
  const int col = TILE_COL(n0, lane);
  const float bi = bias[col];
  for (int r = 0; r < 8; ++r) {
    const int row = TILE_ROW(m0, lane, r);
    float v = c[r] + bi;
    float g = 0.5f * v * (1.0f + erff(v * 0.70710678118654752f));  // exact gelu
    gout[((size_t)b * Ndim + row) * Hdim + col] = g;
  }
}

// ---------------------------------------------------------------------------
// Persistent LSTM: whole T=24 recurrence in one launch.
// Block = 16 rows x 256 threads (8 waves). Rows never mix across blocks, so
// h/c live in LDS; W_ih/W_hh B-fragments + biases live in registers.
// Per step: 8 waves cover the 16x256 gate matrix (2 col-tiles each), then the
// block applies the cell nonlinearity and swaps state, all inside LDS.
// ---------------------------------------------------------------------------
__global__ void __launch_bounds__(256)
k_lstm(const float* __restrict__ xh, const float* __restrict__ Wih,
       const float* __restrict__ Whh, const float* __restrict__ bih,
       const float* __restrict__ bhh, float* __restrict__ hout) {
  __shared__ float sG[16 * 256];   // gates for this step
  __shared__ float sH[16 * Hdim];  // h state (persistent across t)
  __shared__ float sC[16 * Hdim];  // c state (persistent across t)

  const int tid = threadIdx.x;
  const int lane = tid & 31;
  const int wv = tid >> 5;         // 0..7
  const int m0 = blockIdx.x * 16;  // global row stripe

  // zero-init state
  for (int i = tid; i < 16 * Hdim; i += 256) { sH[i] = 0.0f; sC[i] = 0.0f; }

  // ---- preload B fragments (W^T) and biases into registers ----
  const int kh = (lane >> 4) << 1;
  const int lrow = lane & 15;                  // local A row
  v2f Bih[2][16], Bhh[2][16];
  float gbias[2];
  int gcol[2];
#pragma unroll
  for (int tt = 0; tt < 2; ++tt) {
    const int col = wv * 32 + tt * 16 + (lane & 15);  // gate column 0..255
    gcol[tt] = col;
    gbias[tt] = bih[col] + bhh[col];
#pragma unroll
    for (int kk = 0; kk < 16; ++kk) {
      const int ka = kk * 4 + kh;
      Bih[tt][kk].x = Wih[col * Hdim + ka];
      Bih[tt][kk].y = Wih[col * Hdim + ka + 1];
      Bhh[tt][kk].x = Whh[col * Hdim + ka];
      Bhh[tt][kk].y = Whh[col * Hdim + ka + 1];
    }
  }
  const float* xrow0 = xh + (size_t)(m0 + lrow) * Tdim * Hdim;
  __syncthreads();

  for (int t = 0; t < Tdim; ++t) {
    // ---- gates = x_t @ W_ih^T + h @ W_hh^T  (one A-frag feeds both tiles) ----
    v8f a0 = {}, a1 = {};
    const float* xrow = xrow0 + t * Hdim;
#pragma unroll
    for (int kk = 0; kk < 16; ++kk) {
      const int ka = kk * 4 + kh;
      v2f ax;
      ax.x = xrow[ka];
      ax.y = xrow[ka + 1];
      a0 = wmma4(ax, Bih[0][kk], a0);
      a1 = wmma4(ax, Bih[1][kk], a1);
    }
#pragma unroll
    for (int kk = 0; kk < 16; ++kk) {
      const int ka = kk * 4 + kh;
      v2f ah;
      ah.x = sH[lrow * Hdim + ka];
      ah.y = sH[lrow * Hdim + ka + 1];
      a0 = wmma4(ah, Bhh[0][kk], a0);
      a1 = wmma4(ah, Bhh[1][kk], a1);
    }
    const int rbase = (lane >> 4) * 8;
#pragma unroll
    for (int r = 0; r < 8; ++r) {
      sG[(rbase + r) * 256 + gcol[0]] = a0[r] + gbias[0];
      sG[(rbase + r) * 256 + gcol[1]] = a1[r] + gbias[1];
    }
    __syncthreads();

    // ---- cell nonlinearity: 16 rows x 64 hidden over 256 threads ----
    const int hc = tid & 63;
    for (int rr = tid >> 6; rr < 16; rr += 4) {
      float gi = sG[rr * 256 + hc];
      float gf = sG[rr * 256 + 64 + hc];
      float gg = sG[rr * 256 + 128 + hc];
      float go = sG[rr * 256 + 192 + hc];
      float cp = sC[rr * Hdim + hc];
      float i = 1.0f / (1.0f + expf(-gi));
      float f = 1.0f / (1.0f + expf(-gf));
      float o = 1.0f / (1.0f + expf(-go));
      float cn = f * cp + i * tanhf(gg);
      sC[rr * Hdim + hc] = cn;
      sH[rr * Hdim + hc] = o * tanhf(cn);
    }
    __syncthreads();  // state visible before next step's WMMA reads
  }

  // ---- emit final h ----
  for (int i = tid; i < 16 * Hdim; i += 256)
    hout[(size_t)m0 * Hdim + i] = sH[i];
}

// ---------------------------------------------------------------------------
// Head: out = (LN(g2)*g_g+b_g + LN(l)*g_l+b_l) @ W_d + b_d  per (b,n) row
// ---------------------------------------------------------------------------
__global__ void k_head(const float* __restrict__ g2, const float* __restrict__ lh,
                       const float* __restrict__ gg, const float* __restrict__ bg,
                       const float* __restrict__ gl, const float* __restrict__ bl,
                       const float* __restrict__ Wd, const float* __restrict__ bd,
                       float* __restrict__ out) {
  const int row = blockIdx.x;
  const int j = threadIdx.x;  // 64 threads
  __shared__ float red[64];
  __shared__ float sv[64];
  float a = g2[(size_t)row * Hdim + j];
  float b = lh[(size_t)row * Hdim + j];

#define BLOCK_SUM(val, outv)                                     \
  red[j] = (val);                                                \
  __syncthreads();                                               \
  for (int s = 32; s > 0; s >>= 1) {                             \
    if (j < s) red[j] += red[j + s];                             \
    __syncthreads();                                             \
  }                                                              \
  outv = red[0];                                                 \
  __syncthreads();

  float sum, mua, vara, mub, varb;
  BLOCK_SUM(a, sum);          mua = sum * (1.0f / 64.0f);
  float da = a - mua;
  BLOCK_SUM(da * da, sum);    vara = sum * (1.0f / 64.0f);
  BLOCK_SUM(b, sum);          mub = sum * (1.0f / 64.0f);
  float db = b - mub;
  BLOCK_SUM(db * db, sum);    varb = sum * (1.0f / 64.0f);
#undef BLOCK_SUM

  float v = da * rsqrtf(vara + 1e-5f) * gg[j] + bg[j] +
            db * rsqrtf(varb + 1e-5f) * gl[j] + bl[j];
  sv[j] = v;
  __syncthreads();
  if (j < PRED) {
    float s = bd[j];
    for (int k = 0; k < Hdim; ++k) s += sv[k] * Wd[k * PRED + j];
    out[(size_t)row * PRED + j] = s;
  }
}

// ---------------------------------------------------------------------------
// host launcher
// ---------------------------------------------------------------------------
extern "C" void kernel_launch(void* const* d_in, const int* in_sizes, int n_in,
                              void* d_out, int out_size, void* d_ws, size_t ws_size,
                              hipStream_t stream) {
  const float* x    = (const float*)d_in[0];
  const int*   ei   = (const int*)d_in[1];
  const float* Wfc  = (const float*)d_in[2];
  const float* bfc  = (const float*)d_in[3];
  const float* W1   = (const float*)d_in[4];
  const float* al1  = (const float*)d_in[5];
  const float* ar1  = (const float*)d_in[6];
  const float* bi1  = (const float*)d_in[7];
  const float* W2   = (const float*)d_in[8];
  const float* al2  = (const float*)d_in[9];
  const float* ar2  = (const float*)d_in[10];
  const float* bi2  = (const float*)d_in[11];
  const float* Wih  = (const float*)d_in[12];
  const float* Whh  = (const float*)d_in[13];
  const float* bih  = (const float*)d_in[14];
  const float* bhh  = (const float*)d_in[15];
  const float* g_l  = (const float*)d_in[16];
  const float* b_l  = (const float*)d_in[17];
  const float* g_g  = (const float*)d_in[18];
  const float* b_g  = (const float*)d_in[19];
  const float* Wd   = (const float*)d_in[20];
  const float* bd   = (const float*)d_in[21];
  const int E = in_sizes[1] / 2;

  float* ws    = (float*)d_ws;
  float* adj   = ws;                              // 512*512       = 262144
  float* xh    = adj + Ndim * Ndim;               // 49152*64      = 3145728
  float* Hb    = xh + (size_t)XROWS * Hdim;       // 2048*64
  float* el    = Hb + (size_t)Mrows * Hdim;       // 2048
  float* er    = el + Mrows;                      // 2048
  float* alpha = er + Mrows;                      // 4*512*512     = 1048576
  float* g1    = alpha + (size_t)Bdim * Ndim * Ndim;  // 2048*64
  float* g2    = g1 + (size_t)Mrows * Hdim;       // 2048*64
  float* hb    = g2 + (size_t)Mrows * Hdim;       // 2048*64 (LSTM final h)

  // adjacency additive bias: -inf everywhere, 0 on edges
  k_fill<<<(Ndim * Ndim + 255) / 256, 256, 0, stream>>>(adj, -INFINITY, Ndim * Ndim);
  k_scatter_adj<<<(E + 255) / 256, 256, 0, stream>>>(ei, E, adj);

  // x_h = x @ W_fc + b_fc  (all timesteps; LSTM needs them)
  k_xh<<<1536, 256, 0, stream>>>(x, Wfc, bfc, xh);

  // ---- GAT layer 1 at t = T-1 only (GAT is per-timestep) ----
  k_gat_h<<<64, 256, 0, stream>>>(xh + (Tdim - 1) * Hdim, Tdim * Hdim, W1, Hb);
  k_elr<<<256, 256, 0, stream>>>(Hb, al1, ar1, el, er);
  k_alpha<<<Mrows, 256, 0, stream>>>(el, er, adj, alpha);
  k_gat_out<<<64, 256, 0, stream>>>(alpha, Hb, bi1, g1);

  // ---- GAT layer 2 ----
  k_gat_h<<<64, 256, 0, stream>>>(g1, Hdim, W2, Hb);
  k_elr<<<256, 256, 0, stream>>>(Hb, al2, ar2, el, er);
  k_alpha<<<Mrows, 256, 0, stream>>>(el, er, adj, alpha);
  k_gat_out<<<64, 256, 0, stream>>>(alpha, Hb, bi2, g2);

  // ---- LSTM: single persistent kernel, state in LDS ----
  k_lstm<<<Mrows / 16, 256, 0, stream>>>(xh, Wih, Whh, bih, bhh, hb);

  // ---- LayerNorm fusion + decode ----
  k_head<<<Mrows, 64, 0, stream>>>(g2, hb, g_g, b_g, g_l, b_l, Wd, bd,
                                   (float*)d_out);
}